// VecDecoder_91010357002500
// MI455X (gfx1250) — compile-verified
//
#include <hip/hip_runtime.h>
#include <math.h>

#define B 64
#define S 512
#define H 1024
#define T 64

typedef float v2f __attribute__((ext_vector_type(2)));
typedef float v8f __attribute__((ext_vector_type(8)));

__device__ __forceinline__ float fast_sigmoid(float x) {
    return 1.0f / (1.0f + __expf(-x));
}
__device__ __forceinline__ float fast_tanh(float x) {
    // robust for large |x|: e -> inf gives 1 - 0 = 1
    float e = __expf(2.0f * fabsf(x));
    float t = 1.0f - 2.0f / (e + 1.0f);
    return copysignf(t, x);
}

__global__ __launch_bounds__(256) void fill_kernel(float* __restrict__ p, float v, int n) {
    int i = blockIdx.x * blockDim.x + threadIdx.x;
    if (i < n) p[i] = v;
}

// ---------------------------------------------------------------------------
// Fused GRU cell. Each wave owns one 16x16 tile of h_new.
// C = X * W^T uses V_WMMA_F32_16X16X4_F32 (fp32 in, fp32 acc).
// A (16x4 f32): lane holds M=lane&15, float2 at K = k + 2*(lane>>4).
// B (4x16 = W^T tile): lane holds N=lane&15, float2 from W row (col0+N) at same K.
// C/D (16x16 f32, 8 VGPR): element (M = v + 8*(lane>>4), N = lane&15).
// ---------------------------------------------------------------------------
__global__ __launch_bounds__(128) void gru_wmma_kernel(
    const float* __restrict__ x, int ldx,
    const float* __restrict__ h_in,
    const float* __restrict__ w_ih, const float* __restrict__ w_hh,
    const float* __restrict__ b_ih, const float* __restrict__ b_hh,
    float* __restrict__ h_out) {
    const int lane = threadIdx.x & 31;
    const int wave = threadIdx.x >> 5;
    const int row0 = blockIdx.x * 16;                  // batch tile
    const int col0 = (blockIdx.y * 4 + wave) * 16;     // hidden tile
    const int m  = lane & 15;
    const int kh = (lane >> 4) << 1;                   // 0 or 2

    v8f a_ir = {}, a_iz = {}, a_in = {}, a_hr = {}, a_hz = {}, a_hn = {};

    const float* xr  = x    + (size_t)(row0 + m) * ldx + kh;
    const float* hr  = h_in + (size_t)(row0 + m) * H   + kh;
    const float* wir = w_ih + (size_t)(0 * H + col0 + m) * H + kh;
    const float* wiz = w_ih + (size_t)(1 * H + col0 + m) * H + kh;
    const float* win = w_ih + (size_t)(2 * H + col0 + m) * H + kh;
    const float* whr = w_hh + (size_t)(0 * H + col0 + m) * H + kh;
    const float* whz = w_hh + (size_t)(1 * H + col0 + m) * H + kh;
    const float* whn = w_hh + (size_t)(2 * H + col0 + m) * H + kh;

    for (int k = 0; k < H; k += 4) {
        v2f ax = *(const v2f*)(xr + k);
        v2f ah = *(const v2f*)(hr + k);
        v2f b0 = *(const v2f*)(wir + k);
        v2f b1 = *(const v2f*)(wiz + k);
        v2f b2 = *(const v2f*)(win + k);
        v2f b3 = *(const v2f*)(whr + k);
        v2f b4 = *(const v2f*)(whz + k);
        v2f b5 = *(const v2f*)(whn + k);
        a_ir = __builtin_amdgcn_wmma_f32_16x16x4_f32(false, ax, false, b0, (short)0, a_ir, false, false);
        a_iz = __builtin_amdgcn_wmma_f32_16x16x4_f32(false, ax, false, b1, (short)0, a_iz, false, false);
        a_in = __builtin_amdgcn_wmma_f32_16x16x4_f32(false, ax, false, b2, (short)0, a_in, false, false);
        a_hr = __builtin_amdgcn_wmma_f32_16x16x4_f32(false, ah, false, b3, (short)0, a_hr, false, false);
        a_hz = __builtin_amdgcn_wmma_f32_16x16x4_f32(false, ah, false, b4, (short)0, a_hz, false, false);
        a_hn = __builtin_amdgcn_wmma_f32_16x16x4_f32(false, ah, false, b5, (short)0, a_hn, false, false);
    }

    const int n = lane & 15;
    const float bir = b_ih[0 * H + col0 + n];
    const float biz = b_ih[1 * H + col0 + n];
    const float bin = b_ih[2 * H + col0 + n];
    const float bhr = b_hh[0 * H + col0 + n];
    const float bhz = b_hh[1 * H + col0 + n];
    const float bhn = b_hh[2 * H + col0 + n];
    const int mbase = row0 + ((lane >> 4) << 3);
#pragma unroll
    for (int v = 0; v < 8; ++v) {
        const size_t idx = (size_t)(mbase + v) * H + col0 + n;
        float r  = fast_sigmoid(a_ir[v] + bir + a_hr[v] + bhr);
        float z  = fast_sigmoid(a_iz[v] + biz + a_hz[v] + bhz);
        float nn = fast_tanh(a_in[v] + bin + r * (a_hn[v] + bhn));
        h_out[idx] = (1.0f - z) * nn + z * h_in[idx];
    }
}

// ---------------------------------------------------------------------------
// out = tanh([mix, h] @ w_out^T + b_out). Same WMMA tiling; the K=2048
// reduction is split into the mix half and the h half of the concat.
// ---------------------------------------------------------------------------
__global__ __launch_bounds__(128) void out_wmma_kernel(
    const float* __restrict__ mix, const float* __restrict__ h,
    const float* __restrict__ w_out, const float* __restrict__ b_out,
    float* __restrict__ out, int ldo) {
    const int lane = threadIdx.x & 31;
    const int wave = threadIdx.x >> 5;
    const int row0 = blockIdx.x * 16;
    const int col0 = (blockIdx.y * 4 + wave) * 16;
    const int m  = lane & 15;
    const int kh = (lane >> 4) << 1;

    v8f acc = {};
    const float* mr = mix + (size_t)(row0 + m) * H + kh;
    const float* hp = h   + (size_t)(row0 + m) * H + kh;
    const float* wr = w_out + (size_t)(col0 + m) * (2 * H) + kh;

    for (int k = 0; k < H; k += 4) {   // K = 0..1023 : mix
        v2f a0 = *(const v2f*)(mr + k);
        v2f b0 = *(const v2f*)(wr + k);
        acc = __builtin_amdgcn_wmma_f32_16x16x4_f32(false, a0, false, b0, (short)0, acc, false, false);
    }
    for (int k = 0; k < H; k += 4) {   // K = 1024..2047 : h
        v2f a0 = *(const v2f*)(hp + k);
        v2f b0 = *(const v2f*)(wr + H + k);
        acc = __builtin_amdgcn_wmma_f32_16x16x4_f32(false, a0, false, b0, (short)0, acc, false, false);
    }

    const int n = lane & 15;
    const float bb = b_out[col0 + n];
    const int mbase = row0 + ((lane >> 4) << 3);
#pragma unroll
    for (int v = 0; v < 8; ++v)
        out[(size_t)(mbase + v) * ldo + col0 + n] = fast_tanh(acc[v] + bb);
}

// ---------------------------------------------------------------------------
// scores[b][s] = dot(h[b], E[b][s]); one wave per (b,s), float4 streaming
// loads (L2-resident after first step), wave32 shuffle reduction.
// ---------------------------------------------------------------------------
__global__ __launch_bounds__(256) void scores_kernel(
    const float* __restrict__ h, const float* __restrict__ E,
    float* __restrict__ scores) {
    const int wid  = (int)(blockIdx.x * 8 + (threadIdx.x >> 5));
    const int lane = threadIdx.x & 31;
    const int b = wid >> 9;       // / 512
    const int s = wid & 511;
    const float* hb = h + (size_t)b * H;
    const float* e  = E + ((size_t)b * S + s) * H;
    float acc = 0.0f;
#pragma unroll
    for (int i = lane * 4; i < H; i += 128) {
        float4 ev = *(const float4*)(e + i);
        float4 hv = *(const float4*)(hb + i);
        acc = fmaf(ev.x, hv.x, fmaf(ev.y, hv.y, fmaf(ev.z, hv.z, fmaf(ev.w, hv.w, acc))));
    }
#pragma unroll
    for (int off = 16; off; off >>= 1) acc += __shfl_xor(acc, off, 32);
    if (lane == 0) scores[(size_t)b * S + s] = acc;
}

// Softmax over S=512 per batch row (in place). 256 threads, 2 elems/thread.
__global__ __launch_bounds__(256) void softmax_kernel(float* __restrict__ scores) {
    __shared__ float red[8];
    const int b = blockIdx.x, tid = threadIdx.x;
    float* sc = scores + (size_t)b * S;
    float v0 = sc[tid], v1 = sc[tid + 256];
    float m = fmaxf(v0, v1);
#pragma unroll
    for (int off = 16; off; off >>= 1) m = fmaxf(m, __shfl_xor(m, off, 32));
    if ((tid & 31) == 0) red[tid >> 5] = m;
    __syncthreads();
    float bm = red[0];
#pragma unroll
    for (int i = 1; i < 8; ++i) bm = fmaxf(bm, red[i]);
    __syncthreads();
    float e0 = __expf(v0 - bm), e1 = __expf(v1 - bm);
    float sum = e0 + e1;
#pragma unroll
    for (int off = 16; off; off >>= 1) sum += __shfl_xor(sum, off, 32);
    if ((tid & 31) == 0) red[tid >> 5] = sum;
    __syncthreads();
    float tot = 0.0f;
#pragma unroll
    for (int i = 0; i < 8; ++i) tot += red[i];
    const float inv = 1.0f / tot;
    sc[tid] = e0 * inv;
    sc[tid + 256] = e1 * inv;
}

// mix[b][d] = sum_s attn[b][s] * E[b][s][d]; coalesced across d.
__global__ __launch_bounds__(256) void mix_kernel(
    const float* __restrict__ attn, const float* __restrict__ E,
    float* __restrict__ mix) {
    const int b = blockIdx.x;
    const int d = blockIdx.y * 256 + threadIdx.x;
    const float* Eb = E + (size_t)b * S * H + d;
    const float* ab = attn + (size_t)b * S;
    float acc = 0.0f;
#pragma unroll 4
    for (int s = 0; s < S; ++s)
        acc = fmaf(ab[s], Eb[(size_t)s * H], acc);
    mix[(size_t)b * H + d] = acc;
}

extern "C" void kernel_launch(void* const* d_in, const int* in_sizes, int n_in,
                              void* d_out, int out_size, void* d_ws, size_t ws_size,
                              hipStream_t stream) {
    (void)in_sizes; (void)n_in; (void)out_size;
    const float* enc_h = (const float*)d_in[0];   // (1,B,H)
    /* d_in[1] = lengths: unused by the reference math (softmax is unmasked) */
    const float* E     = (const float*)d_in[2];   // (B,S,H)
    const float* w_ih  = (const float*)d_in[3];   // (3H,H)
    const float* w_hh  = (const float*)d_in[4];   // (3H,H)
    const float* b_ih  = (const float*)d_in[5];
    const float* b_hh  = (const float*)d_in[6];
    const float* w_out = (const float*)d_in[7];   // (H,2H)
    const float* b_out = (const float*)d_in[8];

    float* outputs = (float*)d_out;               // (B,T,H)
    float* hT_out  = outputs + (size_t)B * T * H; // (1,B,H)

    float* ws     = (float*)d_ws;                 // ~1.2 MB used
    float* xinit  = ws;                           // B*H
    float* hbuf0  = xinit + (size_t)B * H;        // B*H
    float* hbuf1  = hbuf0 + (size_t)B * H;        // B*H
    float* scores = hbuf1 + (size_t)B * H;        // B*S
    float* mix    = scores + (size_t)B * S;       // B*H
    (void)ws_size;

    fill_kernel<<<(B * H + 255) / 256, 256, 0, stream>>>(xinit, 0.1f, B * H);

    float* hb[2] = {hbuf0, hbuf1};
    for (int t = 0; t < T; ++t) {
        // decoder input: step t reads the attended output of step t-1
        const float* x  = (t == 0) ? xinit : (outputs + (size_t)(t - 1) * H);
        const int   ldx = (t == 0) ? H : T * H;
        const float* h_in  = (t == 0) ? enc_h : hb[(t - 1) & 1];
        float*       h_out = hb[t & 1];

        gru_wmma_kernel<<<dim3(B / 16, H / 64), 128, 0, stream>>>(
            x, ldx, h_in, w_ih, w_hh, b_ih, b_hh, h_out);

        scores_kernel<<<(B * S) / 8, 256, 0, stream>>>(h_out, E, scores);
        softmax_kernel<<<B, 256, 0, stream>>>(scores);
        mix_kernel<<<dim3(B, H / 256), 256, 0, stream>>>(scores, E, mix);

        out_wmma_kernel<<<dim3(B / 16, H / 64), 128, 0, stream>>>(
            mix, h_out, w_out, b_out, outputs + (size_t)t * H, T * H);
    }

    // decoder_hidden output = final h (step T-1 wrote hb[(T-1)&1] = hbuf1)
    hipMemcpyAsync(hT_out, hb[(T - 1) & 1], (size_t)B * H * sizeof(float),
                   hipMemcpyDeviceToDevice, stream);
}